// PerceptualLoss_30528627540030
// MI455X (gfx1250) — compile-verified
//
#include <hip/hip_runtime.h>
#include <cstdint>

// ---------------------------------------------------------------------------
// PerceptualLoss (SSIM + edge-SSIM + color hist + blockwise color distance)
// for MI455X / gfx1250.  The 11x11 separable Gaussian convolutions (the only
// compute-heavy part) run on the matrix pipe via V_WMMA_F32_16X16X4_F32
// (exact f32 -> no precision loss in the E[x^2]-mu^2 cancellation).
// Banded Gaussian operand slices are pre-baked per (step,lane) into an LDS
// table in WMMA operand layout -> one ds_load_b64 per step, no exec masking.
// Workspace use: ~50 KB of d_ws.
// ---------------------------------------------------------------------------

typedef __attribute__((ext_vector_type(2))) float v2f;
typedef __attribute__((ext_vector_type(8))) float v8f;

#define HW   262144   // 512*512
#define CHW  786432   // 3*512*512

// ws layout (32-bit slots):
//  0 pmax(maxkey) 1 pmin(minkey) 2 tmax 3 tmin
//  4 epmax 5 epmin 6 etmax 7 etmin
//  8 ssim_sum(f) 9 edge_ssim_sum(f) 10 blockdist_sum(f) 11..15 pad
//  16..39 histP(u32)  40..63 histT(u32)
//  64..6207 pred block means (f)   6208..12351 target block means (f)

__device__ __forceinline__ v8f wmma4(v2f a, v2f b, v8f c) {
  // D = A(16x4,f32) x B(4x16,f32) + C(16x16,f32)
  return __builtin_amdgcn_wmma_f32_16x16x4_f32(false, a, false, b, (short)0, c,
                                               false, false);
}

// order-preserving float<->uint key for atomic min/max on any-sign floats
__device__ __forceinline__ unsigned fkey(float f) {
  unsigned u = __float_as_uint(f);
  return (u & 0x80000000u) ? ~u : (u | 0x80000000u);
}
__device__ __forceinline__ float unfkey(unsigned k) {
  return (k & 0x80000000u) ? __uint_as_float(k ^ 0x80000000u)
                           : __uint_as_float(~k);
}

__device__ __forceinline__ float wredSum(float v) {
  for (int o = 16; o; o >>= 1) v += __shfl_down(v, o, 32);
  return v;
}
__device__ __forceinline__ float wredMax(float v) {
  for (int o = 16; o; o >>= 1) v = fmaxf(v, __shfl_down(v, o, 32));
  return v;
}
__device__ __forceinline__ float wredMin(float v) {
  for (int o = 16; o; o >>= 1) v = fminf(v, __shfl_down(v, o, 32));
  return v;
}

// normalized 11-tap Gaussian coefficient, g(d)=0 outside [0,10]
__device__ __forceinline__ float gausd(int d) {
  if ((unsigned)d > 10u) return 0.f;
  float x = (float)(d - 5);
  // normalization constant: sum_{i=0..10} exp(-(i-5)^2/4.5)
  float s = 0.f;
#pragma unroll
  for (int i = 0; i < 11; ++i) {
    float t = (float)(i - 5);
    s += __expf(-t * t / 4.5f);
  }
  return __expf(-x * x / 4.5f) / s;
}

// ---------------------------------------------------------------------------
__global__ void init_kernel(unsigned* ws_u, float* ws_f) {
  int t = threadIdx.x;
  if (t < 8)        ws_u[t] = (t & 1) ? 0xFFFFFFFFu : 0u; // even=max,odd=min
  else if (t < 16)  ws_f[t] = 0.f;
  else if (t < 64)  ws_u[t] = 0u;
}

// ---------------------------------------------------------------------------
// Pass 1: global min/max of pred & target + per-channel 8-bin histograms.
__global__ __launch_bounds__(256)
void minmax_hist_kernel(const float* __restrict__ P, const float* __restrict__ T,
                        unsigned* ws_u) {
  __shared__ unsigned hist[48];
  __shared__ float red[32];
  for (int i = threadIdx.x; i < 48; i += 256) hist[i] = 0u;
  __syncthreads();

  float pmx = -3.4e38f, pmn = 3.4e38f, tmx = -3.4e38f, tmn = 3.4e38f;
  const size_t N = (size_t)25165824;
  for (size_t i = (size_t)blockIdx.x * 256 + threadIdx.x; i < N;
       i += (size_t)gridDim.x * 256) {
    float pv = P[i], tv = T[i];
    pmx = fmaxf(pmx, pv); pmn = fminf(pmn, pv);
    tmx = fmaxf(tmx, tv); tmn = fminf(tmn, tv);
    int c = (int)((i >> 18) % 3u);
    int pb = (int)floorf(pv * 8.f); pb = pb < 0 ? 0 : (pb > 7 ? 7 : pb);
    int tb = (int)floorf(tv * 8.f); tb = tb < 0 ? 0 : (tb > 7 ? 7 : tb);
    atomicAdd(&hist[c * 8 + pb], 1u);
    atomicAdd(&hist[24 + c * 8 + tb], 1u);
  }
  pmx = wredMax(pmx); pmn = wredMin(pmn);
  tmx = wredMax(tmx); tmn = wredMin(tmn);
  int wave = threadIdx.x >> 5, lane = threadIdx.x & 31;
  if (lane == 0) { red[wave] = pmx; red[8+wave] = pmn; red[16+wave] = tmx; red[24+wave] = tmn; }
  __syncthreads();
  if (threadIdx.x == 0) {
    float a = red[0], b = red[8], c = red[16], d = red[24];
    for (int w = 1; w < 8; ++w) {
      a = fmaxf(a, red[w]);     b = fminf(b, red[8+w]);
      c = fmaxf(c, red[16+w]);  d = fminf(d, red[24+w]);
    }
    atomicMax(&ws_u[0], fkey(a)); atomicMin(&ws_u[1], fkey(b));
    atomicMax(&ws_u[2], fkey(c)); atomicMin(&ws_u[3], fkey(d));
  }
  if (threadIdx.x < 48) atomicAdd(&ws_u[16 + threadIdx.x], hist[threadIdx.x]);
}

// ---------------------------------------------------------------------------
// Pass 2: min/max of the Sobel-edge maps (gray + zero-padded 3x3, on the fly).
__device__ __forceinline__ float gray0(const float* __restrict__ B3, int y, int x) {
  if ((unsigned)y > 511u || (unsigned)x > 511u) return 0.f;
  int o = (y << 9) + x;
  return (B3[o] + B3[o + HW] + B3[o + 2 * HW]) * (1.f / 3.f);
}

__global__ __launch_bounds__(256)
void edge_minmax_kernel(const float* __restrict__ P, const float* __restrict__ T,
                        unsigned* ws_u) {
  int i = blockIdx.x * 256 + threadIdx.x;            // exactly 32*512*512
  int b = i >> 18; int r = i & (HW - 1); int y = r >> 9; int x = r & 511;
  const float* Pb = P + (size_t)b * CHW;
  const float* Tb = T + (size_t)b * CHW;
  // K = sx+sy = [[2,2,0],[2,0,-2],[0,-2,-2]]
  float ep = 2.f*(gray0(Pb,y-1,x-1)+gray0(Pb,y-1,x)+gray0(Pb,y,x-1))
           - 2.f*(gray0(Pb,y,x+1)+gray0(Pb,y+1,x)+gray0(Pb,y+1,x+1));
  float et = 2.f*(gray0(Tb,y-1,x-1)+gray0(Tb,y-1,x)+gray0(Tb,y,x-1))
           - 2.f*(gray0(Tb,y,x+1)+gray0(Tb,y+1,x)+gray0(Tb,y+1,x+1));
  float a0 = wredMax(ep), a1 = wredMin(ep), a2 = wredMax(et), a3 = wredMin(et);
  __shared__ float red[32];
  int wave = threadIdx.x >> 5, lane = threadIdx.x & 31;
  if (lane == 0) { red[wave]=a0; red[8+wave]=a1; red[16+wave]=a2; red[24+wave]=a3; }
  __syncthreads();
  if (threadIdx.x == 0) {
    float m0=red[0], m1=red[8], m2=red[16], m3=red[24];
    for (int w = 1; w < 8; ++w) {
      m0=fmaxf(m0,red[w]); m1=fminf(m1,red[8+w]);
      m2=fmaxf(m2,red[16+w]); m3=fminf(m3,red[24+w]);
    }
    atomicMax(&ws_u[4], fkey(m0)); atomicMin(&ws_u[5], fkey(m1));
    atomicMax(&ws_u[6], fkey(m2)); atomicMin(&ws_u[7], fkey(m3));
  }
}

// ---------------------------------------------------------------------------
// Pass 3: 64x64 block means per (b,c); 6144 workgroups.
__global__ __launch_bounds__(256)
void block_mean_kernel(const float* __restrict__ P, const float* __restrict__ T,
                       float* ws_f) {
  int bid = blockIdx.x;
  int bx = bid & 7, by = (bid >> 3) & 7, c = (bid >> 6) % 3, b = bid / 192;
  const float* Pb = P + ((size_t)b * 3 + c) * HW;
  const float* Tb = T + ((size_t)b * 3 + c) * HW;
  float sp = 0.f, st = 0.f;
  for (int k = 0; k < 16; ++k) {
    int q = threadIdx.x + 256 * k;
    int yy = q >> 6, xx = q & 63;
    int o = ((by * 64 + yy) << 9) + (bx * 64 + xx);
    sp += Pb[o]; st += Tb[o];
  }
  sp = wredSum(sp); st = wredSum(st);
  __shared__ float red[16];
  int wave = threadIdx.x >> 5, lane = threadIdx.x & 31;
  if (lane == 0) { red[wave] = sp; red[8 + wave] = st; }
  __syncthreads();
  if (threadIdx.x == 0) {
    float a = 0.f, b2 = 0.f;
    for (int w = 0; w < 8; ++w) { a += red[w]; b2 += red[8 + w]; }
    ws_f[64 + bid]   = a  * (1.f / 4096.f);
    ws_f[6208 + bid] = b2 * (1.f / 4096.f);
  }
}

// Pass 4: per-block L2-over-channels distance, summed.
__global__ __launch_bounds__(256)
void block_dist_kernel(float* ws_f) {
  int id = blockIdx.x * 256 + threadIdx.x;           // 2048 total
  int b = id >> 6, r = id & 63;
  float d2 = 0.f;
  for (int c = 0; c < 3; ++c) {
    int o = 64 + ((b * 3 + c) << 6) + r;
    float d = ws_f[o] - ws_f[o + 6144];
    d2 += d * d;
  }
  float s = wredSum(sqrtf(d2));
  __shared__ float red[8];
  int wave = threadIdx.x >> 5, lane = threadIdx.x & 31;
  if (lane == 0) red[wave] = s;
  __syncthreads();
  if (threadIdx.x == 0) {
    float a = 0.f;
    for (int w = 0; w < 8; ++w) a += red[w];
    atomicAdd(&ws_f[10], a);
  }
}

// ---------------------------------------------------------------------------
// SSIM via WMMA f32 16x16x4. One wave = one 16x16 output tile.
// Separable 11-tap Gaussian: H = IN(26x28) x Wh(28x16) [2 chains],
// OUT = Wv(16x28) x H(28x16) [1 chain], each chain = 7 K=4 WMMA steps.
// The banded constant operand (Wh for horizontal-B, Wv for vertical-A) has
// the identical per-(step,lane) values in both layouts -> one shared LDS
// table wtab[7][32] of float2, fetched with a single ds_load_b64 per step.
template <bool EDGE>
__global__ __launch_bounds__(128)
void ssim_wmma_kernel(const float* __restrict__ P, const float* __restrict__ T,
                      float* ws_f, const unsigned* __restrict__ ws_u,
                      int sumSlot, int mmBase) {
  __shared__ v2f  wtab[7 * 32];   // coefficient slices in WMMA operand layout
  __shared__ float lds[4][3776];  // per wave: win_p(728)+win_t(728)+union(2320)

  // build the coefficient table once per block (224 entries, 128 threads)
  for (int idx = threadIdx.x; idx < 7 * 32; idx += 128) {
    int s = idx >> 5, l = idx & 31;
    int d0 = 4 * s + 2 * (l >> 4) - (l & 15);
    v2f w; w.x = gausd(d0); w.y = gausd(d0 + 1);
    wtab[idx] = w;
  }
  __syncthreads();

  const int wave = threadIdx.x >> 5, lane = threadIdx.x & 31;
  const int half = lane >> 4, lm = lane & 15;
  float* Wn   = lds[wave];
  float* winP = Wn;            // [26][28]
  float* winT = Wn + 728;      // [26][28]
  float* Hl   = Wn + 1456;     // 5 maps x [16 cols][stride 29]; aliases gray buf

  const int tid = blockIdx.x * 4 + wave;
  const int tx = tid & 31, ty = (tid >> 5) & 31;
  int b, ch;
  if (EDGE) { ch = 0; b = tid >> 10; }
  else      { ch = (tid >> 10) % 3; b = tid / 3072; }
  (void)ch;
  const int ox = tx * 16, oy = ty * 16;

  const float dr = fmaxf(unfkey(ws_u[mmBase + 0]) - unfkey(ws_u[mmBase + 1]),
                         unfkey(ws_u[mmBase + 2]) - unfkey(ws_u[mmBase + 3]));
  const float c1 = (0.01f * dr) * (0.01f * dr);
  const float c2 = (0.03f * dr) * (0.03f * dr);

  // ---- stage raw p/t 26x28 windows into LDS -------------------------------
  if (EDGE) {
    float* gP = Hl;            // [28][30]
    float* gT = Hl + 840;
    const float* Pb = P + (size_t)b * CHW;
    const float* Tb = T + (size_t)b * CHW;
    for (int idx = lane; idx < 840; idx += 32) {
      int r = idx / 30, j = idx - r * 30;
      int y = oy + r - 1, x = ox + j - 1;
      float vp = 0.f, vt = 0.f;
      if (y >= 0 && y < 512 && x >= 0 && x < 512) {
        int o = (y << 9) + x;
        vp = (Pb[o] + Pb[o + HW] + Pb[o + 2 * HW]) * (1.f / 3.f);
        vt = (Tb[o] + Tb[o + HW] + Tb[o + 2 * HW]) * (1.f / 3.f);
      }
      gP[idx] = vp; gT[idx] = vt;
    }
    __syncthreads();
    for (int idx = lane; idx < 728; idx += 32) {
      int rr = idx / 28, jj = idx - rr * 28;
      int a0 = rr * 30 + jj;
      winP[idx] = 2.f*(gP[a0]+gP[a0+1]+gP[a0+30]) - 2.f*(gP[a0+32]+gP[a0+61]+gP[a0+62]);
      winT[idx] = 2.f*(gT[a0]+gT[a0+1]+gT[a0+30]) - 2.f*(gT[a0+32]+gT[a0+61]+gT[a0+62]);
    }
  } else {
    const float* Pb = P + ((size_t)b * 3 + ch) * HW;
    const float* Tb = T + ((size_t)b * 3 + ch) * HW;
    for (int idx = lane; idx < 728; idx += 32) {
      int r = idx / 28, j = idx - r * 28;
      int y = oy + r; if (y > 511) y = 511;
      int x = ox + j; if (x > 511) x = 511;
      int o = (y << 9) + x;
      winP[idx] = Pb[o]; winT[idx] = Tb[o];
    }
  }
  __syncthreads();

  // ---- horizontal conv: 2 chains (rows 0-15, 16-25), 5 maps ---------------
  for (int h = 0; h < 2; ++h) {
    v8f aP = {0,0,0,0,0,0,0,0}, aT = aP, aPP = aP, aTT = aP, aPT = aP;
    int row = lm + 16 * h; if (row > 25) row = 25;
    const int rb = row * 28;
    for (int s = 0; s < 7; ++s) {
      const int c0 = 4 * s + 2 * half;
      v2f p2, t2;
      p2.x = winP[rb + c0]; p2.y = winP[rb + c0 + 1];
      t2.x = winT[rb + c0]; t2.y = winT[rb + c0 + 1];
      const v2f bh = wtab[s * 32 + lane];   // Wh[j][c]=g[j-c] slice, B layout
      aP  = wmma4(p2,      bh, aP);
      aT  = wmma4(t2,      bh, aT);
      aPP = wmma4(p2 * p2, bh, aPP);
      aTT = wmma4(t2 * t2, bh, aTT);
      aPT = wmma4(p2 * t2, bh, aPT);
    }
#pragma unroll
    for (int v = 0; v < 8; ++v) {            // D layout -> column-major H in LDS
      const int M = 16 * h + v + 8 * half;
      if (M < 28) {
        const int base = lm * 29 + M;
        Hl[0 * 464 + base] = aP[v];
        Hl[1 * 464 + base] = aT[v];
        Hl[2 * 464 + base] = aPP[v];
        Hl[3 * 464 + base] = aTT[v];
        Hl[4 * 464 + base] = aPT[v];
      }
    }
  }
  __syncthreads();

  // ---- vertical conv: OUT = Wv x H, 5 maps --------------------------------
  v8f cP = {0,0,0,0,0,0,0,0}, cT = cP, cPP = cP, cTT = cP, cPT = cP;
  for (int s = 0; s < 7; ++s) {
    const int c0 = 4 * s + 2 * half;
    const v2f av = wtab[s * 32 + lane];      // Wv[r][j]=g[j-r] slice, A layout
    const int hb = lm * 29 + c0;
    v2f bb;
    bb.x = Hl[0*464+hb]; bb.y = Hl[0*464+hb+1]; cP  = wmma4(av, bb, cP);
    bb.x = Hl[1*464+hb]; bb.y = Hl[1*464+hb+1]; cT  = wmma4(av, bb, cT);
    bb.x = Hl[2*464+hb]; bb.y = Hl[2*464+hb+1]; cPP = wmma4(av, bb, cPP);
    bb.x = Hl[3*464+hb]; bb.y = Hl[3*464+hb+1]; cTT = wmma4(av, bb, cTT);
    bb.x = Hl[4*464+hb]; bb.y = Hl[4*464+hb+1]; cPT = wmma4(av, bb, cPT);
  }

  // ---- pointwise SSIM + masked accumulation -------------------------------
  float local = 0.f;
#pragma unroll
  for (int v = 0; v < 8; ++v) {
    const float mup = cP[v], mut = cT[v];
    const float mpp = mup * mup, mtt = mut * mut, mpt = mup * mut;
    const float num = (2.f * mpt + c1) * (2.f * (cPT[v] - mpt) + c2);
    const float den = (mpp + mtt + c1) * ((cPP[v] - mpp) + (cTT[v] - mtt) + c2);
    const int yy = oy + v + 8 * half, xx = ox + lm;
    if (yy < 502 && xx < 502) local += num / den;
  }
  local = wredSum(local);
  if (lane == 0) atomicAdd(&ws_f[sumSlot], local);
}

// ---------------------------------------------------------------------------
__global__ void finalize_kernel(const unsigned* __restrict__ ws_u,
                                const float* __restrict__ ws_f, float* out) {
  float hd2 = 0.f;
  for (int c = 0; c < 3; ++c) {
    float sp = 0.f, st = 0.f;
    for (int k = 0; k < 8; ++k) {
      sp += (float)ws_u[16 + c * 8 + k];
      st += (float)ws_u[40 + c * 8 + k];
    }
    float ip = 1.f / (sp + 1e-8f), it = 1.f / (st + 1e-8f);
    for (int k = 0; k < 8; ++k) {
      float d = (float)ws_u[16 + c * 8 + k] * ip - (float)ws_u[40 + c * 8 + k] * it;
      hd2 += d * d;
    }
  }
  const float color_reward = 1.f - sqrtf(hd2);
  const float ssim  = ws_f[8] / (32.f * 3.f * 502.f * 502.f);
  const float essim = ws_f[9] / (32.f * 502.f * 502.f);
  const float cdist = 1.f - ws_f[10] * (1.f / (64.f * 32.f));
  out[0] = 0.25f * ssim + 0.05f * essim + 0.25f * color_reward + 0.45f * cdist;
}

// ---------------------------------------------------------------------------
extern "C" void kernel_launch(void* const* d_in, const int* in_sizes, int n_in,
                              void* d_out, int out_size, void* d_ws, size_t ws_size,
                              hipStream_t stream) {
  (void)in_sizes; (void)n_in; (void)out_size; (void)ws_size;
  const float* P = (const float*)d_in[0];
  const float* T = (const float*)d_in[1];
  float*    ws_f = (float*)d_ws;
  unsigned* ws_u = (unsigned*)d_ws;
  float*    out  = (float*)d_out;

  init_kernel<<<1, 64, 0, stream>>>(ws_u, ws_f);
  minmax_hist_kernel<<<4096, 256, 0, stream>>>(P, T, ws_u);
  edge_minmax_kernel<<<32768, 256, 0, stream>>>(P, T, ws_u);   // 32*512*512 threads
  block_mean_kernel<<<6144, 256, 0, stream>>>(P, T, ws_f);
  block_dist_kernel<<<8, 256, 0, stream>>>(ws_f);
  // 32 batches * {3,1} channels * 32*32 tiles, 4 waves (tiles) per block
  ssim_wmma_kernel<false><<<24576, 128, 0, stream>>>(P, T, ws_f, ws_u, 8, 0);
  ssim_wmma_kernel<true ><<< 8192, 128, 0, stream>>>(P, T, ws_f, ws_u, 9, 4);
  finalize_kernel<<<1, 1, 0, stream>>>(ws_u, ws_f, out);
}